// Attention_85873576116895
// MI455X (gfx1250) — compile-verified
//
#include <hip/hip_runtime.h>
#include <stdint.h>

// ---------------------------------------------------------------------------
// MaxViT-style attention block on MI455X (gfx1250), fp32 via V_WMMA_F32_16X16X4_F32
// Global->LDS staging via GLOBAL_LOAD_ASYNC_TO_LDS_B128 (ASYNCcnt).
// ---------------------------------------------------------------------------

typedef float v2f __attribute__((ext_vector_type(2)));
typedef float v8f __attribute__((ext_vector_type(8)));

__device__ __forceinline__ v8f wmma_f32(v2f a, v2f b, v8f c) {
  // D(16x16) += A(16x4) * B(4x16), fp32.
  return __builtin_amdgcn_wmma_f32_16x16x4_f32(
      /*neg_a=*/false, a, /*neg_b=*/false, b,
      /*c_mod=*/(short)0, c, /*reuse_a=*/false, /*reuse_b=*/false);
}

__device__ __forceinline__ v8f zero8() {
  v8f z;
#pragma unroll
  for (int j = 0; j < 8; ++j) z[j] = 0.0f;
  return z;
}

// Async copy 16B from global to LDS (per-lane addresses). Generic LDS pointer:
// low 32 bits of the flat address are the LDS byte offset (ISA aperture rule).
__device__ __forceinline__ void async_g2l_b128(void* lds_dst, const void* gsrc) {
  unsigned loff = (unsigned)(size_t)lds_dst;
  asm volatile("global_load_async_to_lds_b128 %0, %1, off"
               :: "v"(loff), "v"(gsrc)
               : "memory");
}
__device__ __forceinline__ void async_wait0() {
  asm volatile("s_wait_asynccnt 0x0" ::: "memory");
}

// ---------------------------------------------------------------------------
// GroupNorm(1,C): per-batch mean/var over a contiguous chunk of epb elements.
// Deterministic two-stage tree reduction (no float atomics).
// ---------------------------------------------------------------------------
__global__ __launch_bounds__(256) void gn_partial_k(const float* __restrict__ src,
                                                    long long epb, int pbb,
                                                    float* __restrict__ partials) {
  int blk = blockIdx.x;
  int b = blk / pbb;
  int p = blk % pbb;
  long long chunk = epb / pbb;
  const float* ptr = src + (long long)b * epb + (long long)p * chunk;
  float s = 0.f, s2 = 0.f;
  for (long long i = threadIdx.x; i < chunk; i += 256) {
    float v = ptr[i];
    s += v;
    s2 += v * v;
  }
  __shared__ float r1[256], r2[256];
  r1[threadIdx.x] = s;
  r2[threadIdx.x] = s2;
  __syncthreads();
  for (int st = 128; st > 0; st >>= 1) {
    if ((int)threadIdx.x < st) {
      r1[threadIdx.x] += r1[threadIdx.x + st];
      r2[threadIdx.x] += r2[threadIdx.x + st];
    }
    __syncthreads();
  }
  if (threadIdx.x == 0) {
    partials[blk * 2 + 0] = r1[0];
    partials[blk * 2 + 1] = r2[0];
  }
}

__global__ __launch_bounds__(256) void gn_finalize_k(const float* __restrict__ partials,
                                                     int pbb, long long epb, float eps,
                                                     float* __restrict__ stats) {
  int b = blockIdx.x;
  __shared__ float r1[256], r2[256];
  float s = 0.f, s2 = 0.f;
  for (int i = threadIdx.x; i < pbb; i += 256) {
    s += partials[(b * pbb + i) * 2 + 0];
    s2 += partials[(b * pbb + i) * 2 + 1];
  }
  r1[threadIdx.x] = s;
  r2[threadIdx.x] = s2;
  __syncthreads();
  for (int st = 128; st > 0; st >>= 1) {
    if ((int)threadIdx.x < st) {
      r1[threadIdx.x] += r1[threadIdx.x + st];
      r2[threadIdx.x] += r2[threadIdx.x + st];
    }
    __syncthreads();
  }
  if (threadIdx.x == 0) {
    float mean = r1[0] / (float)epb;
    float var = r2[0] / (float)epb - mean * mean;
    stats[b * 2 + 0] = mean;
    stats[b * 2 + 1] = rsqrtf(var + eps);
  }
}

// Normalize BCHW x -> token-major xn[(b*4096+s)*256 + c]
__global__ __launch_bounds__(256) void gn_apply_x_k(const float* __restrict__ x,
                                                    const float* __restrict__ stats,
                                                    const float* __restrict__ w,
                                                    const float* __restrict__ bv,
                                                    float* __restrict__ xn) {
  size_t idx = (size_t)blockIdx.x * 256 + threadIdx.x;  // 4*256*4096 = 2^22 total
  int b = (int)(idx >> 20);
  int rem = (int)(idx & 0xFFFFFu);
  int c = rem >> 12;
  int s = rem & 4095;
  float mean = stats[b * 2], rs = stats[b * 2 + 1];
  xn[((size_t)b * 4096 + s) * 256 + c] = (x[idx] - mean) * rs * w[c] + bv[c];
}

// y_tok = x(BCHW) + gx(token-major)
__global__ __launch_bounds__(256) void resid_k(const float* __restrict__ x,
                                               const float* __restrict__ gx,
                                               float* __restrict__ y) {
  size_t idx = (size_t)blockIdx.x * 256 + threadIdx.x;
  int b = (int)(idx >> 20);
  int rem = (int)(idx & 0xFFFFFu);
  int c = rem >> 12;
  int s = rem & 4095;
  size_t ti = ((size_t)b * 4096 + s) * 256 + c;
  y[ti] = x[idx] + gx[ti];
}

// Normalize token-major src -> token-major dst (may alias)
__global__ __launch_bounds__(256) void gn_apply_tok_k(const float* __restrict__ src,
                                                      const float* __restrict__ stats,
                                                      const float* __restrict__ w,
                                                      const float* __restrict__ bv,
                                                      float* __restrict__ dst,
                                                      int tokpb, int Cc) {
  size_t idx = (size_t)blockIdx.x * 256 + threadIdx.x;
  size_t per = (size_t)tokpb * (size_t)Cc;
  int b = (int)(idx / per);
  int c = (int)(idx % (size_t)Cc);
  float mean = stats[b * 2], rs = stats[b * 2 + 1];
  dst[idx] = (src[idx] - mean) * rs * w[c] + bv[c];
}

// ---------------------------------------------------------------------------
// WMMA fp32 GEMM with LDS staging: C[M,N] = (A1 (+A2))[M,K] @ W[N,K]^T (+bias)
// Block: 256 threads = 8 waves; tile 128(M) x 64(N) x 32(K-chunk).
// A-tile / B-tile staged with global_load_async_to_lds_b128 (ASYNCcnt).
// OUT_MODE 0: row-major C[m*N+n]; 1: BCHW scatter C[(b*N+n)*tok_pb+s].
// ---------------------------------------------------------------------------
template <int HAS_A2, int OUT_MODE>
__global__ __launch_bounds__(256) void gemm_wmma_k(const float* __restrict__ A1,
                                                   const float* __restrict__ A2,
                                                   const float* __restrict__ W,
                                                   const float* __restrict__ bias,
                                                   float* __restrict__ C,
                                                   int M, int N, int K, int tok_pb) {
  __shared__ float As[128][36];  // row pitch 144B (16B aligned)
  __shared__ float Bs[64][36];
  int tid = threadIdx.x;
  int wv = tid >> 5;
  int lane = tid & 31;
  int half = lane >> 4;
  int l16 = lane & 15;
  int m0 = blockIdx.y * 128;
  int n0 = blockIdx.x * 64;

  v8f acc[4];
#pragma unroll
  for (int i = 0; i < 4; ++i) acc[i] = zero8();

  for (int k0 = 0; k0 < K; k0 += 32) {
    // Stage A tile: 128 rows x 32 cols = 1024 float4 (4 per thread)
#pragma unroll
    for (int i = tid; i < 1024; i += 256) {
      int r = i >> 3;
      int cq = (i & 7) * 4;
      const float* src = &A1[(size_t)(m0 + r) * K + k0 + cq];
      if (HAS_A2) {
        float4 v = *(const float4*)src;
        float4 v2 = *(const float4*)&A2[(size_t)(m0 + r) * K + k0 + cq];
        v.x += v2.x; v.y += v2.y; v.z += v2.z; v.w += v2.w;
        *(float4*)&As[r][cq] = v;
      } else {
        async_g2l_b128(&As[r][cq], src);
      }
    }
    // Stage B tile: Bs[n][k] = W[(n0+n)*K + k0 + k]; 512 float4 (2 per thread)
#pragma unroll
    for (int i = tid; i < 512; i += 256) {
      int r = i >> 3;
      int cq = (i & 7) * 4;
      async_g2l_b128(&Bs[r][cq], &W[(size_t)(n0 + r) * K + k0 + cq]);
    }
    async_wait0();
    __syncthreads();

    int mw = wv * 16 + l16;  // this wave's A rows within tile
#pragma unroll
    for (int kk = 0; kk < 32; kk += 4) {
      int kc = kk + 2 * half;
      v2f a;
      a.x = As[mw][kc];
      a.y = As[mw][kc + 1];
#pragma unroll
      for (int nt = 0; nt < 4; ++nt) {
        v2f bb;
        bb.x = Bs[nt * 16 + l16][kc];
        bb.y = Bs[nt * 16 + l16][kc + 1];
        acc[nt] = wmma_f32(a, bb, acc[nt]);
      }
    }
    __syncthreads();
  }

#pragma unroll
  for (int nt = 0; nt < 4; ++nt) {
    int n = n0 + nt * 16 + l16;
    float bvv = bias ? bias[n] : 0.0f;
#pragma unroll
    for (int j = 0; j < 8; ++j) {
      int m = m0 + wv * 16 + j + 8 * half;
      float v = acc[nt][j] + bvv;
      if (OUT_MODE == 0) {
        C[(size_t)m * N + n] = v;
      } else {
        int b = m / tok_pb;
        int s = m % tok_pb;
        C[((size_t)b * N + n) * tok_pb + s] = v;
      }
    }
  }
}

// ---------------------------------------------------------------------------
// Local windowed attention: one block per (b, head, 8x8 window).
// q,k,v: [64,32] from qkv[N,768]; S=QK^T (wmma), softmax (scale applied
// at softmax time), O=PV (wmma). q/k/v staged with async b128 loads.
// ---------------------------------------------------------------------------
__global__ __launch_bounds__(256) void local_attn_k(const float* __restrict__ qkv,
                                                    float* __restrict__ gx) {
  int pid = blockIdx.x;  // b*512 + head*64 + gy*8 + gxw
  int b = pid >> 9;
  int head = (pid >> 6) & 7;
  int win = pid & 63;
  int gy = win >> 3, gxw = win & 7;

  __shared__ float Qs[64][36], Ks[64][36], Vs[64][36], Ps[64][66];
  int tid = threadIdx.x;
  const float scale = 0.17677669529663687f;  // 32^-0.5

  // 512 float4 per matrix; 2 per thread per matrix
#pragma unroll
  for (int i = tid; i < 512; i += 256) {
    int p = i >> 3;
    int d4 = (i & 7) * 4;
    int iy = p >> 3, ix = p & 7;
    int t = b * 4096 + (gy * 8 + iy) * 64 + (gxw * 8 + ix);
    size_t base = (size_t)t * 768 + head * 32 + d4;
    async_g2l_b128(&Qs[p][d4], &qkv[base]);
    async_g2l_b128(&Ks[p][d4], &qkv[base + 256]);
    async_g2l_b128(&Vs[p][d4], &qkv[base + 512]);
  }
  async_wait0();
  __syncthreads();

  int wv = tid >> 5, lane = tid & 31, half = lane >> 4, l16 = lane & 15;

  // S = Q @ K^T : 4x4 tiles of 16x16, 2 per wave
#pragma unroll
  for (int u = 0; u < 2; ++u) {
    int lt = wv * 2 + u;
    int mi = lt >> 2, ni = lt & 3;
    v8f acc = zero8();
    int m = mi * 16 + l16, n = ni * 16 + l16;
#pragma unroll
    for (int k0 = 0; k0 < 32; k0 += 4) {
      int kk = k0 + 2 * half;
      v2f a; a.x = Qs[m][kk]; a.y = Qs[m][kk + 1];
      v2f bb; bb.x = Ks[n][kk]; bb.y = Ks[n][kk + 1];  // B[k,n]=K[n,k]
      acc = wmma_f32(a, bb, acc);
    }
#pragma unroll
    for (int j = 0; j < 8; ++j) Ps[mi * 16 + j + 8 * half][ni * 16 + l16] = acc[j];
  }
  __syncthreads();

  if (tid < 64) {
    float mx = -1e30f;
    for (int j = 0; j < 64; ++j) mx = fmaxf(mx, Ps[tid][j] * scale);
    float s = 0.f;
    for (int j = 0; j < 64; ++j) {
      float e = __expf(Ps[tid][j] * scale - mx);
      Ps[tid][j] = e;
      s += e;
    }
    float inv = 1.0f / s;
    for (int j = 0; j < 64; ++j) Ps[tid][j] *= inv;
  }
  __syncthreads();

  // O = P @ V : 4x2 tiles, one per wave
  {
    int mi = wv >> 1, ni = wv & 1;
    v8f acc = zero8();
    int m = mi * 16 + l16, n = ni * 16 + l16;
#pragma unroll
    for (int k0 = 0; k0 < 64; k0 += 4) {
      int kk = k0 + 2 * half;
      v2f a; a.x = Ps[m][kk]; a.y = Ps[m][kk + 1];
      v2f bb; bb.x = Vs[kk][n]; bb.y = Vs[kk + 1][n];
      acc = wmma_f32(a, bb, acc);
    }
#pragma unroll
    for (int j = 0; j < 8; ++j) {
      int p = mi * 16 + j + 8 * half;
      int d = ni * 16 + l16;
      int iy = p >> 3, ix = p & 7;
      int t = b * 4096 + (gy * 8 + iy) * 64 + (gxw * 8 + ix);
      gx[(size_t)t * 256 + head * 32 + d] = acc[j];
    }
  }
}

// ---------------------------------------------------------------------------
// Patch-merge LayerNorm: one block per pooled token; 4C=1024 values.
// ---------------------------------------------------------------------------
__global__ __launch_bounds__(256) void patch_ln_k(const float* __restrict__ gridx,
                                                  const float* __restrict__ lnw,
                                                  const float* __restrict__ lnb,
                                                  float* __restrict__ mnorm) {
  int tok = blockIdx.x;  // b*1024 + oh*32 + ow
  int b = tok >> 10;
  int s = tok & 1023;
  int oh = s >> 5, ow = s & 31;
  __shared__ float vals[1024];
  __shared__ float r1[256], r2[256];
  int tid = threadIdx.x;
  float sum = 0.f, sum2 = 0.f;
  for (int j = tid; j < 1024; j += 256) {
    int part = j >> 8, c = j & 255;
    int dy = part & 1, dx = part >> 1;  // concat order: (0,0),(1,0),(0,1),(1,1)
    int t = b * 4096 + (2 * oh + dy) * 64 + (2 * ow + dx);
    float v = gridx[(size_t)t * 256 + c];
    vals[j] = v;
    sum += v;
    sum2 += v * v;
  }
  r1[tid] = sum;
  r2[tid] = sum2;
  __syncthreads();
  for (int st = 128; st > 0; st >>= 1) {
    if (tid < st) {
      r1[tid] += r1[tid + st];
      r2[tid] += r2[tid + st];
    }
    __syncthreads();
  }
  float mu = r1[0] / 1024.f;
  float var = r2[0] / 1024.f - mu * mu;
  float rs = rsqrtf(var + 1e-5f);
  for (int j = tid; j < 1024; j += 256)
    mnorm[(size_t)tok * 1024 + j] = (vals[j] - mu) * rs * lnw[j] + lnb[j];
}

// ---------------------------------------------------------------------------
// Global attention (flash-style): one block per (b, head, 64-query tile).
// q:[64,32] vs kv:[1024,32] in 16 key blocks with online softmax.
// K/V tiles staged with async b128 loads each iteration.
// ---------------------------------------------------------------------------
__global__ __launch_bounds__(256) void global_attn_k(const float* __restrict__ qg,
                                                     const float* __restrict__ kvg,
                                                     float* __restrict__ go) {
  int pid = blockIdx.x;  // b*512 + head*64 + qt
  int b = pid >> 9;
  int head = (pid >> 6) & 7;
  int qt = pid & 63;

  __shared__ float Qs[64][36], Ks[64][36], Vs[64][36], Ps[64][66], Or[64][36];
  __shared__ float mrow[64], lrow[64], arow[64];
  int tid = threadIdx.x;
  const float scale = 0.17677669529663687f;

#pragma unroll
  for (int i = tid; i < 512; i += 256) {
    int m = i >> 3;
    int d4 = (i & 7) * 4;
    int t = b * 4096 + qt * 64 + m;
    async_g2l_b128(&Qs[m][d4], &qg[(size_t)t * 256 + head * 32 + d4]);
  }
  for (int idx = tid; idx < 64 * 32; idx += 256) {
    int m = idx >> 5, d = idx & 31;
    Or[m][d] = 0.0f;
  }
  if (tid < 64) {
    mrow[tid] = -1e30f;
    lrow[tid] = 0.0f;
  }
  async_wait0();
  __syncthreads();

  int wv = tid >> 5, lane = tid & 31, half = lane >> 4, l16 = lane & 15;

  for (int kb = 0; kb < 16; ++kb) {
#pragma unroll
    for (int i = tid; i < 512; i += 256) {
      int p = i >> 3;
      int d4 = (i & 7) * 4;
      size_t base = ((size_t)(b * 1024 + kb * 64 + p)) * 512 + head * 32 + d4;
      async_g2l_b128(&Ks[p][d4], &kvg[base]);
      async_g2l_b128(&Vs[p][d4], &kvg[base + 256]);
    }
    async_wait0();
    __syncthreads();

#pragma unroll
    for (int u = 0; u < 2; ++u) {
      int lt = wv * 2 + u;
      int mi = lt >> 2, ni = lt & 3;
      v8f acc = zero8();
      int m = mi * 16 + l16, n = ni * 16 + l16;
#pragma unroll
      for (int k0 = 0; k0 < 32; k0 += 4) {
        int kk = k0 + 2 * half;
        v2f a; a.x = Qs[m][kk]; a.y = Qs[m][kk + 1];
        v2f bb; bb.x = Ks[n][kk]; bb.y = Ks[n][kk + 1];
        acc = wmma_f32(a, bb, acc);
      }
#pragma unroll
      for (int j = 0; j < 8; ++j) Ps[mi * 16 + j + 8 * half][ni * 16 + l16] = acc[j];
    }
    __syncthreads();

    if (tid < 64) {
      float mo = mrow[tid];
      float mb = -1e30f;
      for (int j = 0; j < 64; ++j) mb = fmaxf(mb, Ps[tid][j] * scale);
      float mn = fmaxf(mo, mb);
      float al = __expf(mo - mn);
      float s = 0.f;
      for (int j = 0; j < 64; ++j) {
        float e = __expf(Ps[tid][j] * scale - mn);
        Ps[tid][j] = e;
        s += e;
      }
      lrow[tid] = lrow[tid] * al + s;
      arow[tid] = al;
      mrow[tid] = mn;
    }
    __syncthreads();

    {
      int mi = wv >> 1, ni = wv & 1;
      v8f acc = zero8();
      int m = mi * 16 + l16, n = ni * 16 + l16;
#pragma unroll
      for (int k0 = 0; k0 < 64; k0 += 4) {
        int kk = k0 + 2 * half;
        v2f a; a.x = Ps[m][kk]; a.y = Ps[m][kk + 1];
        v2f bb; bb.x = Vs[kk][n]; bb.y = Vs[kk + 1][n];
        acc = wmma_f32(a, bb, acc);
      }
#pragma unroll
      for (int j = 0; j < 8; ++j) {
        int mm = mi * 16 + j + 8 * half;
        int nn = ni * 16 + l16;
        Or[mm][nn] = Or[mm][nn] * arow[mm] + acc[j];
      }
    }
    __syncthreads();
  }

  for (int idx = tid; idx < 64 * 32; idx += 256) {
    int m = idx >> 5, d = idx & 31;
    int t = b * 4096 + qt * 64 + m;
    go[(size_t)t * 256 + head * 32 + d] = Or[m][d] / lrow[m];
  }
}

// ---------------------------------------------------------------------------
// Host-side orchestration
// ---------------------------------------------------------------------------
extern "C" void kernel_launch(void* const* d_in, const int* in_sizes, int n_in,
                              void* d_out, int out_size, void* d_ws, size_t ws_size,
                              hipStream_t stream) {
  const float* x        = (const float*)d_in[0];
  const float* norm_w   = (const float*)d_in[1];
  const float* norm_b   = (const float*)d_in[2];
  const float* qkv_w    = (const float*)d_in[3];
  const float* qkv_b    = (const float*)d_in[4];
  const float* proj_w   = (const float*)d_in[5];
  const float* proj_b   = (const float*)d_in[6];
  const float* grid_nw  = (const float*)d_in[7];
  const float* grid_nb  = (const float*)d_in[8];
  const float* pm_ln_w  = (const float*)d_in[9];
  const float* pm_ln_b  = (const float*)d_in[10];
  const float* pm_red_w = (const float*)d_in[11];
  const float* ds_nw    = (const float*)d_in[12];
  const float* ds_nb    = (const float*)d_in[13];
  const float* q_w      = (const float*)d_in[14];
  const float* q_b      = (const float*)d_in[15];
  const float* kv_w     = (const float*)d_in[16];
  const float* kv_b     = (const float*)d_in[17];
  float* out = (float*)d_out;

  float* ws = (float*)d_ws;
  float* xn       = ws + 0;                 // 4,194,304  (reused: mnorm, go)
  float* qkv      = ws + 4194304;           // 12,582,912 (reused: qg/kvg/pm/ds)
  float* gx       = ws + 16777216;          // 4,194,304
  float* gridx    = ws + 20971520;          // 4,194,304
  float* partials = ws + 25165824;          // 1024
  float* stats    = partials + 1024;        // 8
  float* mnorm = xn;
  float* go    = xn;
  float* qg    = qkv;
  float* kvg   = qkv + 4194304;
  float* pm    = qkv + 6291456;
  float* ds    = qkv + 7340032;

  dim3 blk(256);

  // 1) GroupNorm(x) -> xn (token-major)
  gn_partial_k<<<4 * 64, blk, 0, stream>>>(x, 1048576LL, 64, partials);
  gn_finalize_k<<<4, blk, 0, stream>>>(partials, 64, 1048576LL, 1e-6f, stats);
  gn_apply_x_k<<<16384, blk, 0, stream>>>(x, stats, norm_w, norm_b, xn);

  // 2) QKV projection: [16384,768] = xn @ qkv_w^T + b
  gemm_wmma_k<0, 0><<<dim3(12, 128), blk, 0, stream>>>(xn, nullptr, qkv_w, qkv_b, qkv,
                                                       16384, 768, 256, 0);

  // 3) Local windowed attention -> gx (token-major)
  local_attn_k<<<2048, blk, 0, stream>>>(qkv, gx);

  // 4) grid_x = GroupNorm(x + gx)
  resid_k<<<16384, blk, 0, stream>>>(x, gx, gridx);
  gn_partial_k<<<4 * 64, blk, 0, stream>>>(gridx, 1048576LL, 64, partials);
  gn_finalize_k<<<4, blk, 0, stream>>>(partials, 64, 1048576LL, 1e-6f, stats);
  gn_apply_tok_k<<<16384, blk, 0, stream>>>(gridx, stats, grid_nw, grid_nb, gridx, 4096, 256);

  // 5) Patch-merge LayerNorm + reduction: pm[4096,256] = mnorm @ pm_red_w^T
  patch_ln_k<<<4096, blk, 0, stream>>>(gridx, pm_ln_w, pm_ln_b, mnorm);
  gemm_wmma_k<0, 0><<<dim3(4, 32), blk, 0, stream>>>(mnorm, nullptr, pm_red_w, nullptr, pm,
                                                     4096, 256, 1024, 0);

  // 6) ds = GroupNorm(pm)
  gn_partial_k<<<4 * 16, blk, 0, stream>>>(pm, 262144LL, 16, partials);
  gn_finalize_k<<<4, blk, 0, stream>>>(partials, 16, 262144LL, 1e-6f, stats);
  gn_apply_tok_k<<<4096, blk, 0, stream>>>(pm, stats, ds_nw, ds_nb, ds, 1024, 256);

  // 7) q / kv projections
  gemm_wmma_k<0, 0><<<dim3(4, 128), blk, 0, stream>>>(gridx, nullptr, q_w, q_b, qg,
                                                      16384, 256, 256, 0);
  gemm_wmma_k<0, 0><<<dim3(8, 32), blk, 0, stream>>>(ds, nullptr, kv_w, kv_b, kvg,
                                                     4096, 512, 256, 0);

  // 8) Global attention (flash) -> go
  global_attn_k<<<2048, blk, 0, stream>>>(qg, kvg, go);

  // 9) out = conv1x1(go + grid_x, proj_w, proj_b), BCHW scatter store
  gemm_wmma_k<1, 1><<<dim3(4, 128), blk, 0, stream>>>(go, gridx, proj_w, proj_b, out,
                                                      16384, 256, 256, 4096);
}